// RelationalGCNLayer_82858509074624
// MI455X (gfx1250) — compile-verified
//
#include <hip/hip_runtime.h>
#include <hip/hip_bf16.h>

typedef __attribute__((ext_vector_type(2))) float v2f;
typedef __attribute__((ext_vector_type(8))) float v8f;

#define F      128     // F_IN == UNITS == 128
#define NREL   8
#define TM     64      // rows of x per block in GEMM
#define XS_PITCH 132   // 128 + 4 pad -> conflict-free A-fragment reads

// ---------------------------------------------------------------------------
// Kernel 1: W_sum = sum_r W[r]  (128x128), b_sum = sum_r b[r]  (128)
// ---------------------------------------------------------------------------
__global__ void k_reduce_wb(const float* __restrict__ W, const float* __restrict__ b,
                            float* __restrict__ Wsum, float* __restrict__ bsum) {
    int idx = blockIdx.x * blockDim.x + threadIdx.x;
    if (idx < F * F) {
        float s = 0.f;
#pragma unroll
        for (int r = 0; r < NREL; ++r) s += W[r * F * F + idx];
        Wsum[idx] = s;
    } else if (idx < F * F + F) {
        int j = idx - F * F;
        float s = 0.f;
#pragma unroll
        for (int r = 0; r < NREL; ++r) s += b[r * F + j];
        bsum[j] = s;
    }
}

// ---------------------------------------------------------------------------
// Kernel 2: zero output accumulator (d_out is poisoned by the harness)
// ---------------------------------------------------------------------------
__global__ void k_zero(float4* __restrict__ out, int n4) {
    int i = blockIdx.x * blockDim.x + threadIdx.x;
    if (i < n4) out[i] = make_float4(0.f, 0.f, 0.f, 0.f);
}

// ---------------------------------------------------------------------------
// Kernel 3: h = x @ W_sum + b_sum via V_WMMA_F32_16X16X4_F32 (full fp32)
//   block = 256 threads = 8 waves; block covers 64 rows, wave w covers
//   N-tile [16w, 16w+16); per wave: 4 M-tiles x 32 K-steps of WMMA.
// ---------------------------------------------------------------------------
__global__ __launch_bounds__(256) void k_gemm(const float* __restrict__ x,
                                              const float* __restrict__ Wsum,
                                              const float* __restrict__ bsum,
                                              float* __restrict__ h,
                                              int n_nodes) {
    __shared__ float xs[TM * XS_PITCH];

    const int tid  = threadIdx.x;
    const int wave = tid >> 5;
    const int lane = tid & 31;
    const int m0   = blockIdx.x * TM;

    // hint: W_sum is small & hot — keep it in cache
    __builtin_prefetch(Wsum, 0, 3);

    // stage 64x128 strip of x into LDS (float4, coalesced; 16B-aligned pitch)
    for (int i = tid; i < TM * (F / 4); i += 256) {
        int r  = i >> 5;        // F/4 == 32
        int c4 = i & 31;
        int gm = m0 + r;
        float4 v = make_float4(0.f, 0.f, 0.f, 0.f);
        if (gm < n_nodes) v = *(const float4*)(x + gm * F + c4 * 4);
        *(float4*)(xs + r * XS_PITCH + c4 * 4) = v;
    }
    __syncthreads();

    const int n0    = wave * 16;
    const int nl    = n0 + (lane & 15);        // this lane's N column
    const int khalf = (lane >= 16) ? 2 : 0;    // K sub-pair per ISA A/B layout

    // Preload all B fragments for this wave's N-tile (W_sum is L2-resident)
    v2f bf[32];
#pragma unroll
    for (int kk = 0; kk < 32; ++kk) {
        int k = kk * 4 + khalf;
        bf[kk].x = Wsum[k * F + nl];
        bf[kk].y = Wsum[(k + 1) * F + nl];
    }
    const float bias = bsum[nl];

#pragma unroll
    for (int mt = 0; mt < 4; ++mt) {
        const float* arow = xs + (mt * 16 + (lane & 15)) * XS_PITCH;
        v8f acc;
#pragma unroll
        for (int j = 0; j < 8; ++j) acc[j] = bias;   // bias folded into C

#pragma unroll
        for (int kk = 0; kk < 32; ++kk) {
            int k = kk * 4 + khalf;
            v2f a;
            a.x = arow[k];
            a.y = arow[k + 1];
            // 8 args: (neg_a, A, neg_b, B, c_mod, C, reuse_a, reuse_b)
            acc = __builtin_amdgcn_wmma_f32_16x16x4_f32(
                false, a, false, bf[kk], (short)0, acc, false, false);
        }

        // C/D layout: VGPR j -> row (j | 8*(lane>=16)), col = lane&15
#pragma unroll
        for (int j = 0; j < 8; ++j) {
            int gm = m0 + mt * 16 + j + ((lane >= 16) ? 8 : 0);
            if (gm < n_nodes) h[gm * F + nl] = acc[j];
        }
    }
}

// ---------------------------------------------------------------------------
// Kernel 4: edge scatter — one wave per edge, 32 lanes x float4 = 128 feats,
//   out[row] += val * h[col]   (hardware global_atomic_add_f32)
// ---------------------------------------------------------------------------
__global__ __launch_bounds__(256) void k_scatter(const int* __restrict__ erow,
                                                 const int* __restrict__ ecol,
                                                 const float* __restrict__ evals,
                                                 const float* __restrict__ h,
                                                 float* __restrict__ out,
                                                 int n_edges) {
    const int wave = threadIdx.x >> 5;
    const int lane = threadIdx.x & 31;
    const int e = blockIdx.x * 8 + wave;
    if (e >= n_edges) return;

    const int   col = ecol[e];
    const int   row = erow[e];
    const float v   = evals[e];

    float4 hv = *(const float4*)(h + col * F + lane * 4);
    float* op = out + row * F + lane * 4;
    unsafeAtomicAdd(op + 0, v * hv.x);
    unsafeAtomicAdd(op + 1, v * hv.y);
    unsafeAtomicAdd(op + 2, v * hv.z);
    unsafeAtomicAdd(op + 3, v * hv.w);
}

// ---------------------------------------------------------------------------
// Kernel 5: out = relu(out)
// ---------------------------------------------------------------------------
__global__ void k_relu(float4* __restrict__ out, int n4) {
    int i = blockIdx.x * blockDim.x + threadIdx.x;
    if (i < n4) {
        float4 v = out[i];
        v.x = fmaxf(v.x, 0.f);
        v.y = fmaxf(v.y, 0.f);
        v.z = fmaxf(v.z, 0.f);
        v.w = fmaxf(v.w, 0.f);
        out[i] = v;
    }
}

// ---------------------------------------------------------------------------
extern "C" void kernel_launch(void* const* d_in, const int* in_sizes, int n_in,
                              void* d_out, int out_size, void* d_ws, size_t ws_size,
                              hipStream_t stream) {
    const float* x     = (const float*)d_in[0];
    const int*   erow  = (const int*)  d_in[1];
    const int*   ecol  = (const int*)  d_in[2];
    const float* evals = (const float*)d_in[3];
    const float* W     = (const float*)d_in[4];
    const float* b     = (const float*)d_in[5];
    float*       out   = (float*)d_out;

    const int n_nodes = in_sizes[0] / F;   // 50000
    const int n_edges = in_sizes[1];       // 400000

    // workspace layout (floats): [0,16384) W_sum | [16384,16512) b_sum |
    //                            [32768, 32768 + n_nodes*F) h
    float* ws   = (float*)d_ws;
    float* Wsum = ws;
    float* bsum = ws + F * F;
    float* h    = ws + 32768;

    const int n4 = (n_nodes * F) / 4;

    k_reduce_wb<<<(F * F + F + 255) / 256, 256, 0, stream>>>(W, b, Wsum, bsum);
    k_zero<<<(n4 + 255) / 256, 256, 0, stream>>>((float4*)out, n4);
    k_gemm<<<(n_nodes + TM - 1) / TM, 256, 0, stream>>>(x, Wsum, bsum, h, n_nodes);
    k_scatter<<<(n_edges + 7) / 8, 256, 0, stream>>>(erow, ecol, evals, h, out, n_edges);
    k_relu<<<(n4 + 255) / 256, 256, 0, stream>>>((float4*)out, n4);
}